// MyModel_61933428415243
// MI455X (gfx1250) — compile-verified
//
#include <hip/hip_runtime.h>
#include <hip/hip_bf16.h>

// Radix-select for the two middle order statistics of 64M floats (8192^2 f32).
// reference outputs: (s_desc[hk], s_desc[hk-1], s_desc[n-hk]) with hk = n/2,
// i.e. ascending ranks hk-1 and hk (and out[2] == out[0]).
// 3 histogram passes (12/12/8 bits) over monotonic u32 keys; both ranks
// tracked simultaneously so HBM traffic is ~3 x 256MB ~= 33us at 23.3 TB/s.

typedef unsigned int u32;

// Monotonic ascending key, branchless: 3 VALU ops (ashr, or, xor).
// neg -> ~bits, pos -> bits | 0x80000000.
__device__ __forceinline__ u32 fkey(float f) {
    u32 u = __float_as_uint(f);
    return u ^ ((u32)(((int)u) >> 31) | 0x80000000u);
}
__device__ __forceinline__ float funkey(u32 k) {
    u32 u = (k & 0x80000000u) ? (k & 0x7fffffffu) : ~k;
    return __uint_as_float(u);
}

// state[0..1] = accumulated key prefixes for rank0/rank1 (bits fixed high->low)
// state[2..3] = remaining rank within the current filtered subset
__global__ void rs_init(u32* __restrict__ state, u32 r0, u32 r1) {
    if (threadIdx.x == 0) {
        state[0] = 0u; state[1] = 0u; state[2] = r0; state[3] = r1;
    }
}

__global__ void rs_zero_hist(u32* __restrict__ hist, int nwords) {
    int i = blockIdx.x * blockDim.x + threadIdx.x;
    if (i < nwords) hist[i] = 0u;
}

// One histogram pass: count `width`-bit digits at `shift` for elements whose
// bits above `topshift` match each rank's current prefix.
// topshift >= 32 (pass 0): prefixes are empty and identical -> build ONE
// histogram and duplicate it at merge time (halves LDS atomic traffic in the
// only pass where every element generates an atomic).
__global__ __launch_bounds__(256) void rs_hist(const float* __restrict__ x,
                                               long long n,
                                               const u32* __restrict__ state,
                                               u32* __restrict__ ghist,
                                               int shift, int nbins, int topshift) {
    extern __shared__ u32 lh[];  // [nbins] (single) or [2*nbins] (dual)
    const bool single = (topshift >= 32);
    const u32 mask = (u32)nbins - 1u;
    const u32 p0 = single ? 0u : (state[0] >> topshift);
    const u32 p1 = single ? 0u : (state[1] >> topshift);

    const int nwords = single ? nbins : 2 * nbins;
    for (int i = threadIdx.x; i < nwords; i += blockDim.x) lh[i] = 0u;
    __syncthreads();

    const float4* __restrict__ x4 = (const float4*)x;
    const long long n4 = n >> 2;  // n = 8192*8192, /4 exact
    const long long stride = (long long)gridDim.x * blockDim.x;

    if (single) {
        for (long long i = (long long)blockIdx.x * blockDim.x + threadIdx.x;
             i < n4; i += stride) {
            if (i + stride < n4) __builtin_prefetch(&x4[i + stride], 0, 1);
            float4 v = x4[i];
            u32 kk[4] = {fkey(v.x), fkey(v.y), fkey(v.z), fkey(v.w)};
#pragma unroll
            for (int c = 0; c < 4; ++c)
                atomicAdd(&lh[(kk[c] >> shift) & mask], 1u);
        }
    } else {
        for (long long i = (long long)blockIdx.x * blockDim.x + threadIdx.x;
             i < n4; i += stride) {
            if (i + stride < n4) __builtin_prefetch(&x4[i + stride], 0, 1);
            float4 v = x4[i];
            u32 kk[4] = {fkey(v.x), fkey(v.y), fkey(v.z), fkey(v.w)};
#pragma unroll
            for (int c = 0; c < 4; ++c) {
                u32 hb = kk[c] >> topshift;
                u32 d = (kk[c] >> shift) & mask;
                if (hb == p0) atomicAdd(&lh[d], 1u);
                if (hb == p1) atomicAdd(&lh[nbins + d], 1u);
            }
        }
    }
    __syncthreads();

    // Sparse merge: normal data touches only a few hundred bins.
    if (single) {
        for (int i = threadIdx.x; i < nbins; i += blockDim.x) {
            u32 c = lh[i];
            if (c) {
                atomicAdd(&ghist[i], c);
                atomicAdd(&ghist[nbins + i], c);
            }
        }
    } else {
        for (int i = threadIdx.x; i < 2 * nbins; i += blockDim.x) {
            u32 c = lh[i];
            if (c) atomicAdd(&ghist[i], c);
        }
    }
}

// Find the bucket containing each remaining rank; fold its digit into the
// prefix. 256 threads pre-sum chunks, thread 0 does the short serial walks.
__global__ __launch_bounds__(256) void rs_scan(u32* __restrict__ state,
                                               const u32* __restrict__ ghist,
                                               int shift, int nbins) {
    __shared__ u32 csum[256];
    const int t = threadIdx.x;
    const int chunk = nbins >> 8;  // nbins is 4096 or 256 -> chunk 16 or 1
    for (int j = 0; j < 2; ++j) {
        const u32* __restrict__ h = ghist + j * nbins;
        u32 s = 0;
        for (int b = t * chunk; b < (t + 1) * chunk; ++b) s += h[b];
        csum[t] = s;
        __syncthreads();
        if (t == 0) {
            u32 r = state[2 + j];
            u32 cum = 0;
            int ci = 0;
            while (ci < 255 && cum + csum[ci] <= r) { cum += csum[ci]; ++ci; }
            int b = ci * chunk;
            while (b < nbins - 1 && cum + h[b] <= r) { cum += h[b]; ++b; }
            state[j] |= ((u32)b) << shift;
            state[2 + j] = r - cum;
        }
        __syncthreads();
    }
}

__global__ void rs_final(const u32* __restrict__ state, float* __restrict__ out) {
    if (threadIdx.x == 0) {
        float v0 = funkey(state[0]);  // ascending rank hk-1 : sort_val, kth_val
        float v1 = funkey(state[1]);  // ascending rank hk   : topk_last
        out[0] = v0;
        out[1] = v1;
        out[2] = v0;
    }
}

extern "C" void kernel_launch(void* const* d_in, const int* in_sizes, int n_in,
                              void* d_out, int out_size, void* d_ws, size_t ws_size,
                              hipStream_t stream) {
    const float* x = (const float*)d_in[0];
    const long long n = (long long)in_sizes[0];   // 8192*8192 = 67108864
    const u32 half_k = (u32)(n / 2);

    u32* state = (u32*)d_ws;       // 4 x u32
    u32* ghist = state + 4;        // up to 2 x 4096 u32 (32KB)
    float* out = (float*)d_out;

    rs_init<<<1, 32, 0, stream>>>(state, half_k - 1u, half_k);

    const int shifts[3] = {20, 8, 0};
    const int widths[3] = {12, 12, 8};
    const int grid = 2048;  // 2048 blocks x 8 wave32s: saturates HBM easily

    for (int p = 0; p < 3; ++p) {
        const int nbins = 1 << widths[p];
        const int shift = shifts[p];
        const int topshift = shift + widths[p];
        const size_t lds = (size_t)((topshift >= 32) ? nbins : 2 * nbins) * sizeof(u32);
        rs_zero_hist<<<(2 * nbins + 255) / 256, 256, 0, stream>>>(ghist, 2 * nbins);
        rs_hist<<<grid, 256, lds, stream>>>(x, n, state, ghist, shift, nbins, topshift);
        rs_scan<<<1, 256, 0, stream>>>(state, ghist, shift, nbins);
    }
    rs_final<<<1, 32, 0, stream>>>(state, out);
}